// RNNTLoss_69939247448572
// MI455X (gfx1250) — compile-verified
//
#include <hip/hip_runtime.h>
#include <hip/hip_bf16.h>
#include <math.h>

// Problem constants (match reference): B=8, T=128, U1=65, V=1024, U=64, BLANK=0
#define RB   8
#define RT   128
#define RU1  65
#define RU   64
#define RV   1024
#define ROWS (RB * RT * RU1)          // 66560 rows of V floats
#define WAVES_PER_BLOCK 8             // 256 threads, wave32

// ---------------------------------------------------------------------------
// Kernel 1: per-row logsumexp over V=1024, emit lp_blank and lp_label.
// One wave (32 lanes) per row. The 4KB row is staged into LDS with CDNA5
// async loads (global_load_async_to_lds_b128, tracked by ASYNCcnt), then
// reduced with wave32 shuffle trees. HBM-bound: 272.6 MB read once.
// ---------------------------------------------------------------------------
__global__ void rnnt_rowlse_kernel(const float* __restrict__ logits,
                                   const int*   __restrict__ y,
                                   float* __restrict__ lp_blank,
                                   float* __restrict__ lp_label) {
    __shared__ float smem[WAVES_PER_BLOCK][RV];   // 32 KB

    const int lane = threadIdx.x & 31;
    const int wave = threadIdx.x >> 5;
    const int row  = blockIdx.x * WAVES_PER_BLOCK + wave;
    if (row >= ROWS) return;

    const float* rowp = logits + (size_t)row * RV;

    // LDS byte offset of this wave's staging buffer (addrspace(3) offset is
    // the low 32 bits of the generic pointer on AMDGPU).
    unsigned lds_base = (unsigned)(size_t)(&smem[wave][0]);

    // 8 chunks: 32 lanes x 16B = 512B per chunk; whole 4KB row async -> LDS.
#pragma unroll
    for (int c = 0; c < 8; ++c) {
        const float* gp = rowp + c * 128 + lane * 4;
        unsigned     lo = lds_base + (unsigned)(c * 512 + lane * 16);
        asm volatile("global_load_async_to_lds_b128 %0, %1, off"
                     :: "v"(lo), "v"(gp) : "memory");
    }
    // Wait for this wave's async copies (per-wave ASYNCcnt).
    asm volatile("s_wait_asynccnt 0x0" ::: "memory");

    // Pull 32 elements/lane from LDS (b128 reads, conflict-free: consecutive
    // lanes hit consecutive 16B), reduce max then sum(exp).
    const float4* sv = reinterpret_cast<const float4*>(&smem[wave][0]);
    float4 x[8];
    float vmax = -INFINITY;
#pragma unroll
    for (int k = 0; k < 8; ++k) {
        x[k] = sv[k * 32 + lane];
        vmax = fmaxf(vmax, fmaxf(fmaxf(x[k].x, x[k].y), fmaxf(x[k].z, x[k].w)));
    }
#pragma unroll
    for (int m = 16; m > 0; m >>= 1)
        vmax = fmaxf(vmax, __shfl_xor(vmax, m, 32));

    float ssum = 0.0f;
#pragma unroll
    for (int k = 0; k < 8; ++k) {
        ssum += __expf(x[k].x - vmax) + __expf(x[k].y - vmax) +
                __expf(x[k].z - vmax) + __expf(x[k].w - vmax);
    }
#pragma unroll
    for (int m = 16; m > 0; m >>= 1)
        ssum += __shfl_xor(ssum, m, 32);

    const float lse = vmax + __logf(ssum);

    if (lane == 0) {
        // blank logit is element 0 of the row (lane0 chunk0 .x)
        lp_blank[row] = x[0].x - lse;
        const int u = row % RU1;
        if (u < RU) {
            const int t = (row / RU1) % RT;
            const int b = row / (RU1 * RT);
            const int lab = y[b * RU + u];            // in [1, V)
            lp_label[((b * RT) + t) * RU + u] = smem[wave][lab] - lse;
        }
    }
}

// ---------------------------------------------------------------------------
// Kernel 2: forward DP (alphas) per batch via anti-diagonal wavefront.
// alpha[t][u] = logaddexp(alpha[t-1][u] + lpb[t-1][u],
//                         alpha[t][u-1] + lpl[t][u-1]),  alpha[0][0]=0.
// One block per batch; lane u <=> column u; 192 diagonals, one barrier each.
// lpb/lpl for this batch live in (dynamic) LDS: 66 KB << 320 KB per WGP.
// ---------------------------------------------------------------------------
__global__ void rnnt_alpha_kernel(const float* __restrict__ lpb_g,
                                  const float* __restrict__ lpl_g,
                                  const int*   __restrict__ logit_lens,
                                  const int*   __restrict__ y_lens,
                                  float* __restrict__ part) {
    extern __shared__ float sdyn[];
    float* s_lpb = sdyn;                 // RT*RU1 = 8320 floats
    float* s_lpl = sdyn + RT * RU1;      // RT*RU  = 8192 floats
    __shared__ float diag[2][RU1 + 1];   // double-buffered anti-diagonal
    __shared__ float res;

    const int b   = blockIdx.x;
    const int tid = threadIdx.x;         // 96 threads (3 waves)

    // Preload this batch's lp tables into LDS (float4, both counts /4).
    {
        const float4* src = (const float4*)(lpb_g + (size_t)b * RT * RU1);
        float4*       dst = (float4*)s_lpb;
        for (int i = tid; i < (RT * RU1) / 4; i += blockDim.x) dst[i] = src[i];
    }
    {
        const float4* src = (const float4*)(lpl_g + (size_t)b * RT * RU);
        float4*       dst = (float4*)s_lpl;
        for (int i = tid; i < (RT * RU) / 4; i += blockDim.x) dst[i] = src[i];
    }
    __syncthreads();

    const int t_end = logit_lens[b] - 1;   // in [63,127]
    const int u_end = y_lens[b];           // in [32,64]
    const int u = tid;
    const float NEG = -INFINITY;
    int p = 0;

    for (int d = 0; d < RT + RU1 - 1; ++d) {   // 192 diagonals
        const int t = d - u;
        const bool active = (u < RU1) && (t >= 0) && (t < RT);
        float val = NEG;
        if (active) {
            if (d == 0) {
                val = 0.0f;                                   // alpha[0][0]
            } else {
                float a  = (t > 0) ? diag[p][u]     + s_lpb[(t - 1) * RU1 + u] : NEG;
                float bb = (u > 0) ? diag[p][u - 1] + s_lpl[t * RU + (u - 1)]  : NEG;
                float m  = fmaxf(a, bb);
                float mn = fminf(a, bb);
                val = (m == NEG) ? NEG : m + log1pf(__expf(mn - m));
            }
            if (t == t_end && u == u_end)
                res = val + s_lpb[t_end * RU1 + u_end];        // log-likelihood
        }
        if (u <= RU1) diag[p ^ 1][u] = val;   // write other buffer (no race)
        __syncthreads();                      // publish diagonal d
        p ^= 1;
    }

    if (tid == 0) part[b] = -res;             // per-batch loss
}

// ---------------------------------------------------------------------------
// Kernel 3: mean over batches -> scalar output (fp32).
// ---------------------------------------------------------------------------
__global__ void rnnt_final_kernel(const float* __restrict__ part,
                                  float* __restrict__ out) {
    if (threadIdx.x == 0 && blockIdx.x == 0) {
        float s = 0.0f;
#pragma unroll
        for (int i = 0; i < RB; ++i) s += part[i];
        out[0] = s * (1.0f / RB);
    }
}

// ---------------------------------------------------------------------------
// Launcher. Inputs (setup_inputs order): logits f32, logit_lens i32, y i32,
// y_lens i32. Output: 1 fp32 scalar. Workspace: lp_blank | lp_label | part.
// ---------------------------------------------------------------------------
extern "C" void kernel_launch(void* const* d_in, const int* in_sizes, int n_in,
                              void* d_out, int out_size, void* d_ws, size_t ws_size,
                              hipStream_t stream) {
    const float* logits     = (const float*)d_in[0];
    const int*   logit_lens = (const int*)d_in[1];
    const int*   y          = (const int*)d_in[2];
    const int*   y_lens     = (const int*)d_in[3];
    float*       out        = (float*)d_out;

    float* ws   = (float*)d_ws;
    float* lpb  = ws;                              // ROWS        = 66560 floats
    float* lpl  = ws + ROWS;                       // RB*RT*RU    = 65536 floats
    float* part = ws + ROWS + RB * RT * RU;        // RB          = 8 floats

    rnnt_rowlse_kernel<<<ROWS / WAVES_PER_BLOCK, 32 * WAVES_PER_BLOCK, 0, stream>>>(
        logits, y, lpb, lpl);

    const size_t dyn_lds = (size_t)(RT * RU1 + RT * RU) * sizeof(float); // 66048 B
    rnnt_alpha_kernel<<<RB, 96, dyn_lds, stream>>>(lpb, lpl, logit_lens, y_lens, part);

    rnnt_final_kernel<<<1, 32, 0, stream>>>(part, out);
}